// GeneralizedCrossEntropy_42975442763791
// MI455X (gfx1250) — compile-verified
//
#include <hip/hip_runtime.h>
#include <hip/hip_bf16.h>
#include <cmath>

typedef __attribute__((ext_vector_type(2))) float        v2f;
typedef __attribute__((ext_vector_type(8))) float        v8f;
typedef __attribute__((ext_vector_type(4))) unsigned int v4u;
typedef __attribute__((ext_vector_type(8))) int          v8i;
typedef __attribute__((ext_vector_type(4))) int          v4i;

namespace {
constexpr int       kB = 16;
constexpr int       kC = 21;
constexpr long long kN = 512LL * 512LL;   // H*W = 262144
constexpr float     kQ = 0.8f;
}

// One workgroup: 16 waves (wave32), wave b owns batch b.
__global__ __launch_bounds__(kB * 32)
void gce_kernel(const float* __restrict__ logits,
                const int*   __restrict__ targets,
                float*       __restrict__ out)
{
    __shared__ float tile[kB * kC * kC];  // [b][c][n] : 28.2 KB, filled by TDM
    __shared__ float ssum[kB * kC];       // softmax denominators S[b][n]
    __shared__ float accv[kC];            // cross-batch accumulator

    const int tid   = (int)threadIdx.x;
    const int lane  = tid & 31;
    const int b     = tid >> 5;           // batch handled by this wave
    const int m     = lane & 15;          // WMMA A-matrix row group
    const int khalf = lane >> 4;          // 0: K+{0,1}, 1: K+{2,3}

    if (tid < kC) accv[tid] = 0.0f;

    // ---------------- TDM: DMA this batch's 21(c) x 21(n) fp32 tile into LDS -------------
    {
        const unsigned long long ga =
            (unsigned long long)(const void*)logits +
            (unsigned long long)b * (unsigned long long)(kC * kN) * 4ull;
        const unsigned int lds_off =
            (unsigned int)(unsigned long long)(const void*)&tile[b * kC * kC];

        v4u g0;
        g0.x = 1u;                                                         // count=1, user mode
        g0.y = (unsigned)__builtin_amdgcn_readfirstlane((int)lds_off);     // lds_addr
        g0.z = (unsigned)__builtin_amdgcn_readfirstlane((int)(unsigned int)ga);        // global_addr[31:0]
        g0.w = (unsigned)__builtin_amdgcn_readfirstlane(
                   (int)(((unsigned int)(ga >> 32) & 0x1FFFFFFu) | (2u << 30)));       // addr[56:32] | type=2

        v8i g1;
        g1[0] = (int)(2u << 16);                                   // workgroup_mask=0, data_size=4B
        g1[1] = (int)(((unsigned)kN & 0xFFFFu) << 16);             // tensor_dim0 low16 (=0)
        g1[2] = (int)(((unsigned)kN >> 16) | ((unsigned)kC << 16)); // tensor_dim0 hi | tensor_dim1 lo
        g1[3] = (int)((unsigned)kC << 16);                          // tensor_dim1 hi(0) | tile_dim0=21
        g1[4] = kC;                                                 // tile_dim1=21 | tile_dim2=0
        g1[5] = (int)kN;                                            // tensor_dim0_stride low32 = 262144
        g1[6] = 0;                                                  // stride hi | dim1_stride lo
        g1[7] = 0;                                                  // dim1_stride hi

        v4i g2 = {0, 0, 0, 0};                                      // 2-D tile: unused dims
        v4i g3 = {0, 0, 0, 0};
        v8i g4 = {0, 0, 0, 0, 0, 0, 0, 0};                          // clang-23 6-arg form: extra group (zero-filled)

        __builtin_amdgcn_tensor_load_to_lds(g0, g1, g2, g3, g4, 0);
    }

    __syncthreads();                        // accv zeros visible to all
    __builtin_amdgcn_s_wait_tensorcnt(0);   // this wave's tile is in LDS

    // ---------------- WMMA: S[n] = sum_c exp(L[c][n]) via E x ones on the matrix pipe ----
    // A (16x4 f32): lane m holds row n=m; VGPR0 = K{0|2}, VGPR1 = K{1|3}.
    // B = ones(4x16)  =>  every column of D = row sums of A-chunk; accumulate over 6 chunks.
    const float* tb = &tile[b * kC * kC];

    #pragma unroll
    for (int mb = 0; mb < 32; mb += 16) {              // rows 0..15, then 16..20 (rest unused)
        v8f c = {0.f, 0.f, 0.f, 0.f, 0.f, 0.f, 0.f, 0.f};
        #pragma unroll
        for (int kb = 0; kb < 24; kb += 4) {           // K chunks cover c = 0..23 (pad -> 0)
            const int row = mb + m;                    // n position
            const int k0  = kb + 2 * khalf;            // channel index for a[0]
            float e0 = 0.0f, e1 = 0.0f;
            if (row < kC) {
                e0 = (k0     < kC) ? expf(tb[ k0      * kC + row]) : 0.0f;
                e1 = (k0 + 1 < kC) ? expf(tb[(k0 + 1) * kC + row]) : 0.0f;
            }
            v2f a;  a[0] = e0;   a[1] = e1;
            v2f bo; bo[0] = 1.0f; bo[1] = 1.0f;
            c = __builtin_amdgcn_wmma_f32_16x16x4_f32(
                    /*neg_a=*/false, a, /*neg_b=*/false, bo,
                    /*c_mod=*/(short)0, c, /*reuse_a=*/false, /*reuse_b=*/false);
        }
        // D layout: lanes<16 hold M=g in c[g]; lanes>=16 hold M=g+8. One lane per half writes.
        if ((lane & 15) == 0) {
            #pragma unroll
            for (int g = 0; g < 8; ++g) {
                const int row = mb + khalf * 8 + g;
                if (row < kC) ssum[b * kC + row] = c[g];
            }
        }
    }

    // ---------------- gather + GCE + cross-batch mean -------------------------------------
    if (lane < kC) {
        const int   k = lane;
        const int   t = targets[(long long)b * kN + k];          // value in [0,21)
        const float p = expf(tb[k * kC + t]) / ssum[b * kC + t]; // softmax prob
        const float part = (1.0f - powf(p, kQ)) * (1.0f / kQ) * (1.0f / (float)kB);
        atomicAdd(&accv[k], part);                               // ds_add_f32
    }
    __syncthreads();
    if (tid < kC) out[tid] = accv[tid];
}

extern "C" void kernel_launch(void* const* d_in, const int* in_sizes, int n_in,
                              void* d_out, int out_size, void* d_ws, size_t ws_size,
                              hipStream_t stream)
{
    (void)in_sizes; (void)n_in; (void)d_ws; (void)ws_size; (void)out_size;
    const float* logits  = (const float*)d_in[0];
    const int*   targets = (const int*)d_in[1];   // harness delivers integer inputs as int32
    float*       out     = (float*)d_out;         // 21 fp32 outputs

    gce_kernel<<<1, kB * 32, 0, stream>>>(logits, targets, out);
}